// LPLayer_single_46402826666668
// MI455X (gfx1250) — compile-verified
//
#include <hip/hip_runtime.h>

#define N_NODES 50000
#define N_EDGES 800000
#define IN_DIM  128
#define OUT_DIM 256

typedef __attribute__((ext_vector_type(2))) float v2f;
typedef __attribute__((ext_vector_type(8))) float v8f;

// ---------------------------------------------------------------- zero ws ---
__global__ void zero_f32(float* __restrict__ p, int n) {
  int i = blockIdx.x * blockDim.x + threadIdx.x;
  if (i < n) p[i] = 0.0f;
}

// ------------------------------------------------------------ edge scatter --
// One wave32 per edge. Each lane moves 4 consecutive floats (32*4 = 128) of
// feat[src] into msg[dst] via coalesced global_atomic_add_f32; lane 0 bumps
// the degree counter. Both feat and msg fit in the 192MB L2, so these atomics
// resolve at L2 without touching HBM.
__global__ void __launch_bounds__(256)
edge_scatter(const float* __restrict__ feat,
             const int* __restrict__ src,
             const int* __restrict__ dst,
             float* __restrict__ msg,
             float* __restrict__ deg) {
  int wid  = (blockIdx.x * blockDim.x + threadIdx.x) >> 5;  // wave32 id
  int lane = threadIdx.x & 31;
  if (wid >= N_EDGES) return;
  int s = src[wid];
  int d = dst[wid];
  const float4 v = *(const float4*)(feat + (size_t)s * IN_DIM + lane * 4);
  float* b = msg + (size_t)d * IN_DIM + lane * 4;
  atomicAdd(b + 0, v.x);
  atomicAdd(b + 1, v.y);
  atomicAdd(b + 2, v.z);
  atomicAdd(b + 3, v.w);
  if (lane == 0) atomicAdd(deg + d, 1.0f);
}

// --------------------------------------------- fused mean + GEMM + ReLU -----
// One wave computes one 16x16 fp32 output tile with V_WMMA_F32_16X16X4_F32,
// chaining 32 WMMAs over K=128. A-fragment elements are computed on the fly:
//   a = 0.5f * (feat[row][k] + msg[row][k] * (1/max(deg,1)))
// A layout (16x4 f32): lane L -> row L%16, K = k + 2*(L/16) + j   (j=0,1)
// B layout (4x16 f32): lane L -> col L%16, same K indexing
// C/D layout: VGPR r -> row r + 8*(L/16), col L%16
__global__ void __launch_bounds__(256)
fused_gemm_relu(const float* __restrict__ feat,
                const float* __restrict__ W,
                const float* __restrict__ msg,
                const float* __restrict__ deg,
                float* __restrict__ out) {
  const int wave = threadIdx.x >> 5;
  const int lane = threadIdx.x & 31;
  const int tile = blockIdx.x * (blockDim.x >> 5) + wave;   // 50000 tiles exact
  const int m0   = (tile >> 4) << 4;   // 3125 row-tiles
  const int n0   = (tile & 15) << 4;   // 16 col-tiles
  const int half = lane >> 4;
  const int mr   = lane & 15;
  const int row  = m0 + mr;            // A row this lane feeds
  const int col  = n0 + mr;            // B column this lane feeds

  const float inv = 1.0f / fmaxf(deg[row], 1.0f);
  const float* __restrict__ frow = feat + (size_t)row * IN_DIM;
  const float* __restrict__ mrow = msg  + (size_t)row * IN_DIM;

  v8f c = {};
  for (int k = 0; k < IN_DIM; k += 4) {
    const int ka = k + 2 * half;
    v2f f = *(const v2f*)(frow + ka);
    v2f m = *(const v2f*)(mrow + ka);
    v2f a;
    a.x = 0.5f * (f.x + m.x * inv);
    a.y = 0.5f * (f.y + m.y * inv);
    v2f b;
    b.x = W[(size_t)ka       * OUT_DIM + col];
    b.y = W[(size_t)(ka + 1) * OUT_DIM + col];
    // 8 args: (neg_a, A, neg_b, B, c_mod, C, reuse_a, reuse_b)
    c = __builtin_amdgcn_wmma_f32_16x16x4_f32(false, a, false, b,
                                              (short)0, c, false, false);
  }

#pragma unroll
  for (int r = 0; r < 8; ++r) {
    const int m = m0 + r + 8 * half;
    out[(size_t)m * OUT_DIM + n0 + mr] = fmaxf(c[r], 0.0f);
  }
}

// ------------------------------------------------------------------ launch --
extern "C" void kernel_launch(void* const* d_in, const int* in_sizes, int n_in,
                              void* d_out, int out_size, void* d_ws, size_t ws_size,
                              hipStream_t stream) {
  const float* feat = (const float*)d_in[0];
  const float* W    = (const float*)d_in[1];
  const int*   src  = (const int*)d_in[2];
  const int*   dst  = (const int*)d_in[3];
  float* out = (float*)d_out;

  float* msg = (float*)d_ws;                       // [N_NODES * IN_DIM]
  float* deg = msg + (size_t)N_NODES * IN_DIM;     // [N_NODES]

  const int nzero = N_NODES * IN_DIM + N_NODES;
  zero_f32<<<(nzero + 255) / 256, 256, 0, stream>>>(msg, nzero);

  const long long nthreads = (long long)N_EDGES * 32;
  edge_scatter<<<(int)((nthreads + 255) / 256), 256, 0, stream>>>(
      feat, src, dst, msg, deg);

  const int tiles = (N_NODES / 16) * (OUT_DIM / 16);   // 50000, 8 waves/block
  fused_gemm_relu<<<tiles / 8, 256, 0, stream>>>(feat, W, msg, deg, out);
}